// DualNN_41094247088345
// MI455X (gfx1250) — compile-verified
//
#include <hip/hip_runtime.h>
#include <hip/hip_bf16.h>

typedef float v2f __attribute__((ext_vector_type(2)));
typedef float v8f __attribute__((ext_vector_type(8)));
typedef int   v4i __attribute__((vector_size(16)));   // matches builtin's expected pointee

#define TILE 64
#define KB 32
#define LDS_STRIDE 36   // 32 + 4 pad: conflict-free column reads, float4-aligned rows

#define AS1 __attribute__((address_space(1)))
#define AS3 __attribute__((address_space(3)))

#if defined(__has_builtin)
#if __has_builtin(__builtin_amdgcn_global_load_async_to_lds_b128)
#define HAVE_ASYNC_LDS 1
#endif
#endif

__device__ __forceinline__ void wait_async_le8() {
#if defined(__has_builtin) && __has_builtin(__builtin_amdgcn_s_wait_asynccnt)
    __builtin_amdgcn_s_wait_asynccnt(8);
#else
    asm volatile("s_wait_asynccnt 0x8" ::: "memory");
#endif
    asm volatile("" ::: "memory");
}
__device__ __forceinline__ void wait_async_0() {
#if defined(__has_builtin) && __has_builtin(__builtin_amdgcn_s_wait_asynccnt)
    __builtin_amdgcn_s_wait_asynccnt(0);
#else
    asm volatile("s_wait_asynccnt 0x0" ::: "memory");
#endif
    asm volatile("" ::: "memory");
}

// C[M x N] = act(A[M x K] @ B[N x K]^T + bias[n]) ; relu applied where global col >= relu_from
// gate != nullptr: uniformly skip whole kernel when *gate == 0 (latched early-exit).
__global__ __launch_bounds__(128)
void gemm_wmma_f32(const float* __restrict__ A, const float* __restrict__ B,
                   const float* __restrict__ bias, float* __restrict__ C,
                   int M, int N, int K, int relu_from, const unsigned* gate)
{
    if (gate && gate[0] == 0u) return;   // uniform across wave -> EXEC all-ones for WMMA

    __shared__ float lA[2][TILE * LDS_STRIDE];
    __shared__ float lB[2][TILE * LDS_STRIDE];

    const int tid  = threadIdx.x;        // 0..127 (4 waves)
    const int wave = tid >> 5;
    const int lane = tid & 31;
    const int wm   = (wave >> 1) * 32;   // wave 2x2 grid over the 64x64 tile
    const int wn   = (wave & 1) * 32;
    const int bm   = blockIdx.y * TILE;
    const int bn   = blockIdx.x * TILE;

    const int lr = lane & 15;            // row-within-16 (A: M, B: N)
    const int kh = (lane >> 4) * 2;      // K offset 0 (lanes 0-15) / 2 (lanes 16-31)

    // Per-thread staging coordinates: 64x32 tile = 512 float4, 4 per thread
    int ofs[4];
    const float* gA[4];
    const float* gB[4];
    #pragma unroll
    for (int i = 0; i < 4; ++i) {
        int idx = tid + i * 128;             // 0..511
        int r   = idx >> 3;                  // 8 float4 per row
        int c   = (idx & 7) * 4;
        ofs[i]  = r * LDS_STRIDE + c;
        gA[i]   = A + (size_t)(bm + r) * K + c;
        gB[i]   = B + (size_t)(bn + r) * K + c;
    }

    const int nk = K / KB;
    v8f acc[2][2] = {};

#ifdef HAVE_ASYNC_LDS
    auto issue_block = [&](int kb2, int buf2) {
        const int k0 = kb2 * KB;
        #pragma unroll
        for (int i = 0; i < 4; ++i) {
            __builtin_amdgcn_global_load_async_to_lds_b128(
                (AS1 v4i*)(gA[i] + k0), (AS3 v4i*)(&lA[buf2][ofs[i]]), 0, 0);
            __builtin_amdgcn_global_load_async_to_lds_b128(
                (AS1 v4i*)(gB[i] + k0), (AS3 v4i*)(&lB[buf2][ofs[i]]), 0, 0);
        }
    };
    issue_block(0, 0);           // prologue: 8 async b128 ops in flight
#endif

    for (int kb = 0; kb < nk; ++kb) {
        const int buf = kb & 1;
#ifdef HAVE_ASYNC_LDS
        if (kb + 1 < nk) {       // prefetch next K-block into the other buffer
            issue_block(kb + 1, buf ^ 1);
            wait_async_le8();    // oldest 8 (this block's) have landed; in-order completion
        } else {
            wait_async_0();
        }
        __syncthreads();
#else
        // Fallback: synchronous stage through VGPRs
        {
            const int k0 = kb * KB;
            #pragma unroll
            for (int i = 0; i < 4; ++i) {
                float4 va = *(const float4*)(gA[i] + k0);
                float4 vb = *(const float4*)(gB[i] + k0);
                *(float4*)(&lA[buf][ofs[i]]) = va;
                *(float4*)(&lB[buf][ofs[i]]) = vb;
            }
        }
        __syncthreads();
#endif
        #pragma unroll
        for (int kk = 0; kk < KB; kk += 4) {
            v2f af[2], bf[2];
            #pragma unroll
            for (int i = 0; i < 2; ++i) {
                // A 16x4 frag: lane -> M=lr, VGPR0/1 -> K = kk+kh / kk+kh+1
                af[i].x = lA[buf][(wm + i * 16 + lr) * LDS_STRIDE + kk + kh];
                af[i].y = lA[buf][(wm + i * 16 + lr) * LDS_STRIDE + kk + kh + 1];
                // B 4x16 frag (B[k,n] = W[n,k]): lane -> N=lr, same K split
                bf[i].x = lB[buf][(wn + i * 16 + lr) * LDS_STRIDE + kk + kh];
                bf[i].y = lB[buf][(wn + i * 16 + lr) * LDS_STRIDE + kk + kh + 1];
            }
            #pragma unroll
            for (int i = 0; i < 2; ++i)
                #pragma unroll
                for (int j = 0; j < 2; ++j)
                    acc[i][j] = __builtin_amdgcn_wmma_f32_16x16x4_f32(
                        false, af[i], false, bf[j], (short)0, acc[i][j], false, false);
        }
        __syncthreads();  // everyone done reading buf before it is overwritten (iter kb+2 issue)
    }

    // Epilogue: C/D layout -> VGPR r: lanes0-15 M=r, lanes16-31 M=r+8; N = lr
    #pragma unroll
    for (int i = 0; i < 2; ++i) {
        #pragma unroll
        for (int j = 0; j < 2; ++j) {
            int row0 = bm + wm + i * 16 + (lane >> 4) * 8;
            int col  = bn + wn + j * 16 + lr;
            float bv = bias ? bias[col] : 0.0f;
            #pragma unroll
            for (int r = 0; r < 8; ++r) {
                float v = acc[i][j][r] + bv;
                if (col >= relu_from) v = fmaxf(v, 0.0f);
                C[(size_t)(row0 + r) * N + col] = v;
            }
        }
    }
}

// bias_proj[n] = sum_m b_vec[m]*WbProj[n,m]; negb[j] = -b_vec[j]; init control block
__global__ void prep_kernel(const float* __restrict__ b_vec, const float* __restrict__ WbProj,
                            float* __restrict__ biasp, float* __restrict__ negb,
                            unsigned* __restrict__ ctrl)
{
    int n = threadIdx.x;                 // 256 threads
    float s = 0.0f;
    #pragma unroll 4
    for (int m = 0; m < 128; ++m) s += b_vec[m] * WbProj[n * 128 + m];
    biasp[n] = s;
    if (n < 128) negb[n] = -b_vec[n];
    if (n == 0) { ctrl[0] = 1u; ctrl[1] = 1u; ctrl[2] = 0x7F800000u; ctrl[3] = 0u; }
}

// evaluate while-loop condition; latch flag off; reset residual accumulator
__global__ void gate_kernel(unsigned* ctrl)
{
    if (ctrl[0]) {
        float res = __uint_as_float(ctrl[2]);
        int   it  = (int)ctrl[1];
        if (!(it <= 20 && res > 1e-6f)) ctrl[0] = 0u;
        else                            ctrl[3] = 0u;
    }
}

__global__ void finish_kernel(unsigned* ctrl)
{
    if (ctrl[0]) { ctrl[2] = ctrl[3]; ctrl[1] = ctrl[1] + 1u; }
}

// grid-strided max|r| -> atomicMax on float bit pattern (non-negative => uint order == float order)
__global__ __launch_bounds__(256)
void absmax_kernel(const float* __restrict__ r, int n, unsigned* ctrl)
{
    if (ctrl[0] == 0u) return;
    float m = 0.0f;
    for (int i = blockIdx.x * blockDim.x + threadIdx.x; i < n; i += gridDim.x * blockDim.x)
        m = fmaxf(m, fabsf(r[i]));
    __shared__ float sm[256];
    sm[threadIdx.x] = m; __syncthreads();
    for (int s = 128; s > 0; s >>= 1) {
        if (threadIdx.x < s) sm[threadIdx.x] = fmaxf(sm[threadIdx.x], sm[threadIdx.x + s]);
        __syncthreads();
    }
    if (threadIdx.x == 0) atomicMax(&ctrl[3], __float_as_uint(sm[0]));
}

// float4 copy, optionally gated
__global__ __launch_bounds__(256)
void copy4_kernel(const float4* __restrict__ src, float4* __restrict__ dst, int n4,
                  const unsigned* gate)
{
    if (gate && gate[0] == 0u) return;
    int i = blockIdx.x * blockDim.x + threadIdx.x;
    if (i < n4) dst[i] = src[i];
}

__global__ void write_iter_kernel(const unsigned* ctrl, float* out)
{
    out[0] = (float)ctrl[1];
}

extern "C" void kernel_launch(void* const* d_in, const int* in_sizes, int n_in,
                              void* d_out, int out_size, void* d_ws, size_t ws_size,
                              hipStream_t stream)
{
    const float* b_primal = (const float*)d_in[0];   // 1024 x 512
    const float* W0       = (const float*)d_in[1];   // 1024 x 512
    const float* b0       = (const float*)d_in[2];   // 1024
    const float* W1       = (const float*)d_in[3];   // 1024 x 1024
    const float* b1       = (const float*)d_in[4];   // 1024
    const float* W2       = (const float*)d_in[5];   // 256 x 1024
    const float* b2       = (const float*)d_in[6];   // 256
    const float* Amat     = (const float*)d_in[7];   // 128 x 256
    const float* b_vec    = (const float*)d_in[8];   // 128
    const float* WzProj   = (const float*)d_in[9];   // 256 x 256
    const float* WbProj   = (const float*)d_in[10];  // 256 x 128

    const int M = 1024, HID = 1024, IN = 512, OUT = 256, MM = 128;
    const int NO_RELU = 1 << 30;

    float* ws = (float*)d_ws;
    float* h1 = ws;                       // 1024*1024
    float* h2 = ws + (1 << 20);           // 1024*1024
    // Iteration-phase buffers alias the (then dead) h1 region:
    float*    zbuf  = ws;                 // 1024*256
    float*    ztmp  = ws + 262144;        // 1024*256
    float*    rbuf  = ws + 524288;        // 1024*128
    float*    biasp = ws + 655360;        // 256
    float*    negb  = biasp + 256;        // 128
    unsigned* ctrl  = (unsigned*)(negb + 128);

    float* z_out   = (float*)d_out;             // 1024*256
    float* out_fwd = (float*)d_out + 262144;    // 1024*256
    float* it_out  = (float*)d_out + 524288;    // 1

    dim3 blk(128);

    // Forward MLP
    gemm_wmma_f32<<<dim3(HID / TILE, M / TILE), blk, 0, stream>>>(
        b_primal, W0, b0, h1, M, HID, IN, 0, nullptr);
    gemm_wmma_f32<<<dim3(HID / TILE, M / TILE), blk, 0, stream>>>(
        h1, W1, b1, h2, M, HID, HID, 0, nullptr);
    gemm_wmma_f32<<<dim3(OUT / TILE, M / TILE), blk, 0, stream>>>(
        h2, W2, b2, out_fwd, M, OUT, HID, NO_RELU, nullptr);

    // bias_proj, -b_vec, control init; z = out
    prep_kernel<<<1, 256, 0, stream>>>(b_vec, WbProj, biasp, negb, ctrl);
    copy4_kernel<<<(262144 / 4) / 256, 256, 0, stream>>>(
        (const float4*)out_fwd, (float4*)zbuf, 262144 / 4, nullptr);

    // Fixed-point loop: fixed 20 trips, device-side latch emulates early exit
    for (int i = 0; i < 20; ++i) {
        gate_kernel<<<1, 1, 0, stream>>>(ctrl);
        // z' = bias_proj + z @ WzProj^T ; relu on cols >= 64
        gemm_wmma_f32<<<dim3(OUT / TILE, M / TILE), blk, 0, stream>>>(
            zbuf, WzProj, biasp, ztmp, M, OUT, OUT, 64, ctrl);
        copy4_kernel<<<(262144 / 4) / 256, 256, 0, stream>>>(
            (const float4*)ztmp, (float4*)zbuf, 262144 / 4, ctrl);
        // r = z @ A^T - b_vec
        gemm_wmma_f32<<<dim3(MM / TILE, M / TILE), blk, 0, stream>>>(
            zbuf, Amat, negb, rbuf, M, MM, OUT, NO_RELU, ctrl);
        absmax_kernel<<<128, 256, 0, stream>>>(rbuf, M * MM, ctrl);
        finish_kernel<<<1, 1, 0, stream>>>(ctrl);
    }

    // Outputs: z_star, out (already in place), curr_iter
    copy4_kernel<<<(262144 / 4) / 256, 256, 0, stream>>>(
        (const float4*)zbuf, (float4*)z_out, 262144 / 4, nullptr);
    write_iter_kernel<<<1, 1, 0, stream>>>(ctrl, it_out);
}